// KIM_23313082483405
// MI455X (gfx1250) — compile-verified
//
#include <hip/hip_runtime.h>

// KIM scorer, algebraically collapsed (size-1 softmaxes == 1, softmax rows sum to 1):
//   score[b] = dot( LN(sum_l news[b,l,:])*g1+b1 , LN(sum_l user[b,l,:])*g2+b2 )
// Memory-bound: 655 MB traffic -> ~28 us floor @ 23.3 TB/s.
// Column sums over L via V_WMMA_F32_16X16X4_F32 (A = ones, exact fp32).
// A wave handles 4 column-tiles at once with an interleaved column mapping
// (tile j owns d = gbase + 4n + j) so each lane feeds all 4 tiles from one
// b128 load -> 4x fewer VMEM instructions + 4 independent WMMA acc chains.

typedef __attribute__((ext_vector_type(2))) float v2f;
typedef __attribute__((ext_vector_type(4))) float v4f;
typedef __attribute__((ext_vector_type(8))) float v8f;

#define B_   4096
#define L_   50
#define D_   400
#define NG_  6       // groups of 64 columns per matrix (6*64 = 384)
#define REM_ 384     // remainder tile covers columns 384..399

__global__ __launch_bounds__(256) void kim_collapsed_score(
    const float* __restrict__ news,   // [B, L, D]
    const float* __restrict__ user,   // [B, L, D]
    const float* __restrict__ g1, const float* __restrict__ be1,
    const float* __restrict__ g2, const float* __restrict__ be2,
    float* __restrict__ out)          // [B]
{
    __shared__ float colsum[2][D_];   // per-batch column sums (news, user)
    __shared__ float rbuf[8][4];
    __shared__ float stats[4];        // mu_n, rstd_n, mu_u, rstd_u

    const int b    = blockIdx.x;
    const int tid  = threadIdx.x;
    const int lane = tid & 31;
    const int wave = tid >> 5;

    const float* base0 = news + (size_t)b * (L_ * D_);
    const float* base1 = user + (size_t)b * (L_ * D_);

    // A = ones(16x4): with all-ones A, D[m,n] = sum_k B[k,n] regardless of
    // the K-slot <-> (VGPR, lane-half) mapping; only n = lane&15 matters.
    v2f a; a.x = 1.0f; a.y = 1.0f;

    const int nsub = lane & 15;          // column-within-tile this lane owns
    const int k0   = (lane >> 4) * 2;    // K rows this lane supplies: k0, k0+1

    const v4f zero4 = {0.f, 0.f, 0.f, 0.f};

    // Work items: 0..11 = 64-column groups (mat = item/6, g = item%6),
    //             12..13 = 16-column remainder tiles (mat = item-12).
    for (int item = wave; item < 2 * NG_ + 2; item += 8) {
        if (item < 2 * NG_) {
            const int   mat   = item / NG_;
            const int   g     = item % NG_;
            const float* X    = (mat == 0) ? base0 : base1;
            const int   gbase = g * 64;
            const int   dvec  = gbase + 4 * nsub;   // this lane's float4 base

            v8f c0 = {}, c1 = {}, c2 = {}, c3 = {};
            #pragma unroll
            for (int l0 = 0; l0 < L_; l0 += 4) {    // 13 chunks; last ragged
                const int lA  = l0 + k0;
                const int lB  = lA + 1;
                const int lAc = (lA < L_) ? lA : (L_ - 1);
                const int lBc = (lB < L_) ? lB : (L_ - 1);
                v4f q0 = *(const v4f*)(X + (size_t)lAc * D_ + dvec);
                v4f q1 = *(const v4f*)(X + (size_t)lBc * D_ + dvec);
                q0 = (lA < L_) ? q0 : zero4;        // v_cndmask, EXEC unchanged
                q1 = (lB < L_) ? q1 : zero4;
                v2f bm0, bm1, bm2, bm3;
                bm0.x = q0[0]; bm0.y = q1[0];
                bm1.x = q0[1]; bm1.y = q1[1];
                bm2.x = q0[2]; bm2.y = q1[2];
                bm3.x = q0[3]; bm3.y = q1[3];
                c0 = __builtin_amdgcn_wmma_f32_16x16x4_f32(false, a, false, bm0, (short)0, c0, false, false);
                c1 = __builtin_amdgcn_wmma_f32_16x16x4_f32(false, a, false, bm1, (short)0, c1, false, false);
                c2 = __builtin_amdgcn_wmma_f32_16x16x4_f32(false, a, false, bm2, (short)0, c2, false, false);
                c3 = __builtin_amdgcn_wmma_f32_16x16x4_f32(false, a, false, bm3, (short)0, c3, false, false);
            }
            // Every row of each D equals the column sum; VGPR0 lanes 0-15 = D[0,n].
            if (lane < 16) {
                colsum[mat][gbase + 4 * lane + 0] = c0[0];
                colsum[mat][gbase + 4 * lane + 1] = c1[0];
                colsum[mat][gbase + 4 * lane + 2] = c2[0];
                colsum[mat][gbase + 4 * lane + 3] = c3[0];
            }
        } else {
            // Remainder tile: columns 384..399, scalar-load path.
            const int   mat = item - 2 * NG_;
            const float* X  = (mat == 0) ? base0 : base1;
            const int   d0  = REM_ + nsub;

            v8f c = {};
            #pragma unroll
            for (int l0 = 0; l0 < L_; l0 += 4) {
                const int lA  = l0 + k0;
                const int lB  = lA + 1;
                const int lAc = (lA < L_) ? lA : (L_ - 1);
                const int lBc = (lB < L_) ? lB : (L_ - 1);
                float vA = X[(size_t)lAc * D_ + d0];
                float vB = X[(size_t)lBc * D_ + d0];
                v2f bm;
                bm.x = (lA < L_) ? vA : 0.0f;
                bm.y = (lB < L_) ? vB : 0.0f;
                c = __builtin_amdgcn_wmma_f32_16x16x4_f32(false, a, false, bm, (short)0, c, false, false);
            }
            if (lane < 16) colsum[mat][REM_ + lane] = c[0];
        }
    }
    __syncthreads();

    // ---- LayerNorm stats for both 400-vectors: Sum(x), Sum(x^2) ----
    float pn = 0.f, pn2 = 0.f, pu = 0.f, pu2 = 0.f;
    for (int d = tid; d < D_; d += 256) {
        float xn = colsum[0][d];
        float xu = colsum[1][d];
        pn += xn; pn2 += xn * xn;
        pu += xu; pu2 += xu * xu;
    }
    #pragma unroll
    for (int off = 16; off > 0; off >>= 1) {
        pn  += __shfl_xor(pn,  off);
        pn2 += __shfl_xor(pn2, off);
        pu  += __shfl_xor(pu,  off);
        pu2 += __shfl_xor(pu2, off);
    }
    if (lane == 0) {
        rbuf[wave][0] = pn;  rbuf[wave][1] = pn2;
        rbuf[wave][2] = pu;  rbuf[wave][3] = pu2;
    }
    __syncthreads();
    if (tid == 0) {
        float sn = 0.f, sn2 = 0.f, su = 0.f, su2 = 0.f;
        #pragma unroll
        for (int w = 0; w < 8; ++w) {
            sn += rbuf[w][0]; sn2 += rbuf[w][1];
            su += rbuf[w][2]; su2 += rbuf[w][3];
        }
        const float inv = 1.0f / (float)D_;
        const float mun = sn * inv, muu = su * inv;
        float varn = fmaxf(sn2 * inv - mun * mun, 0.0f);
        float varu = fmaxf(su2 * inv - muu * muu, 0.0f);
        stats[0] = mun; stats[1] = rsqrtf(varn + 1e-5f);
        stats[2] = muu; stats[3] = rsqrtf(varu + 1e-5f);
    }
    __syncthreads();

    const float mun = stats[0], rn = stats[1];
    const float muu = stats[2], ru = stats[3];

    // ---- dot( LN(news_sum), LN(user_sum) ) ----
    float pdot = 0.f;
    for (int d = tid; d < D_; d += 256) {
        float yn = (colsum[0][d] - mun) * rn * g1[d] + be1[d];
        float yu = (colsum[1][d] - muu) * ru * g2[d] + be2[d];
        pdot += yn * yu;
    }
    #pragma unroll
    for (int off = 16; off > 0; off >>= 1) pdot += __shfl_xor(pdot, off);
    if (lane == 0) rbuf[wave][0] = pdot;
    __syncthreads();
    if (tid == 0) {
        float s = 0.f;
        #pragma unroll
        for (int w = 0; w < 8; ++w) s += rbuf[w][0];
        out[b] = s;
    }
}

extern "C" void kernel_launch(void* const* d_in, const int* in_sizes, int n_in,
                              void* d_out, int out_size, void* d_ws, size_t ws_size,
                              hipStream_t stream) {
    // setup_inputs() order:
    //  0 news_vecs [B,L,D]   1 user_vecs [B,L,D]
    //  2 Wn1  3 bn1  4 Wn2  5 bn2  6 Wu1  7 bu1  8 Wu2  9 bu2  10 Wc  11 bc
    //  (2..11 are dead under the exact size-1-softmax collapse)
    // 12 gamma1 13 beta1 14 gamma2 15 beta2
    const float* news = (const float*)d_in[0];
    const float* user = (const float*)d_in[1];
    const float* g1   = (const float*)d_in[12];
    const float* be1  = (const float*)d_in[13];
    const float* g2   = (const float*)d_in[14];
    const float* be2  = (const float*)d_in[15];
    float* out = (float*)d_out;

    kim_collapsed_score<<<B_, 256, 0, stream>>>(news, user, g1, be1, g2, be2, out);
}